// TemporalDenoise_24232205484033
// MI455X (gfx1250) — compile-verified
//
#include <hip/hip_runtime.h>

// Bidirectional per-channel EMA, B=32 T=2048 C=512 fp32.
// Memory-bound: ~268 MB compulsory traffic @ 23.3 TB/s => ~11.5us roofline.
// 3-kernel chunked affine scan; the fill kernel stages chunks in LDS via the
// gfx1250 async global->LDS path (ASYNCcnt) so x is read once per pass and
// out is written once with non-temporal stores (preserve x in the 192MB L2).

constexpr int Bb  = 32;
constexpr int Tt  = 2048;
constexpr int Cc  = 512;
constexpr int Lc  = 64;          // chunk length along T
constexpr int NCk = Tt / Lc;     // 32 chunks
constexpr int PAD = 1;           // LDS padding (dwords) -> conflict-free

typedef __attribute__((address_space(1))) int   as1_int;
typedef __attribute__((address_space(3))) int   as3_int;
typedef __attribute__((address_space(3))) float as3_float;

__device__ __forceinline__ float sigmoid_f(float z) {
    return 1.0f / (1.0f + __expf(-z));
}

// Async copy of one f32 per lane from global to LDS (gfx1250 path).
__device__ __forceinline__ void async_ld_f32(const float* g, as3_float* l) {
#if defined(__has_builtin) && __has_builtin(__builtin_amdgcn_global_load_async_to_lds_b32)
    __builtin_amdgcn_global_load_async_to_lds_b32((as1_int*)g, (as3_int*)l, 0, 0);
#else
    unsigned loff = (unsigned)(unsigned long long)l;     // LDS byte offset
    unsigned long long ga = (unsigned long long)g;
    asm volatile("global_load_async_to_lds_b32 %0, %1, off"
                 :: "v"(loff), "v"(ga) : "memory");
#endif
}

__device__ __forceinline__ void wait_async0() {
#if defined(__has_builtin) && __has_builtin(__builtin_amdgcn_s_wait_asynccnt)
    __builtin_amdgcn_s_wait_asynccnt(0);
#else
    asm volatile("s_wait_asynccnt 0" ::: "memory");
#endif
    asm volatile("" ::: "memory");   // compiler ordering fence for LDS reads
}

// ---------------- Kernel 3 (listed first for disasm): fused fill ------------
// Per (b,c,chunk): async-load 64-elem chunk to LDS, backward EMA in LDS,
// forward EMA reconstructing a*x[i] = g[i] - w*g[i+1], fused NT store of
// 0.5*(f+g). Carries cf/cb come from kernels 1+2.
__global__ void __launch_bounds__(128) ema_fill(const float* __restrict__ x,
                                                const float* __restrict__ al,
                                                const float* __restrict__ cf,
                                                const float* __restrict__ cb,
                                                float* __restrict__ out) {
    __shared__ float smem[128 * (Lc + PAD)];   // 33,280 B
    const int tid = threadIdx.x;
    const int gid = blockIdx.x * 128 + tid;
    const int c = gid % Cc;
    const int j = (gid / Cc) % NCk;
    const int b = gid / (Cc * NCk);

    const float a = sigmoid_f(al[c]);
    const float w = 1.0f - a;

    const size_t rowbase = ((size_t)(b * Tt + j * Lc)) * Cc + c;
    const float* gp = x + rowbase;
    as3_float* lp = (as3_float*)&smem[tid * (Lc + PAD)];

    // Stage this thread's 64-element chunk into LDS asynchronously.
    // Each instruction: 32 lanes x 4B, contiguous in c => 128B requests.
    #pragma unroll 8
    for (int i = 0; i < Lc; ++i)
        async_ld_f32(gp + (size_t)i * Cc, lp + i);

    const int cidx = (b * NCk + j) * Cc + c;
    const float cfv = cf[cidx];   // forward carry f[jL-1]
    const float cbv = cb[cidx];   // backward carry g[(j+1)L]

    wait_async0();                // own-wave data => no barrier needed

    float* sm = &smem[tid * (Lc + PAD)];

    // Backward EMA within chunk, stored over x in LDS.
    float g = cbv;
    #pragma unroll 4
    for (int i = Lc - 1; i >= 0; --i) {
        g = w * g + a * sm[i];
        sm[i] = g;
    }

    // Forward EMA; out written once, non-temporal (never re-read; keep x in L2).
    float* op = out + rowbase;
    float f = cfv;
    float gcur = sm[0];
    #pragma unroll 4
    for (int i = 0; i < Lc; ++i) {
        const float gnext = (i < Lc - 1) ? sm[i + 1] : cbv;
        f = w * f + (gcur - w * gnext);
        __builtin_nontemporal_store(0.5f * (f + gcur), op + (size_t)i * Cc);
        gcur = gnext;
    }
}

// ---------------- Kernel 1: per-chunk weighted sums -------------------------
// Qf[j] = sum_s a*w^(L-1-s) x[jL+s]   (forward aggregate)
// Qb[j] = sum_s a*w^s       x[jL+s]   (backward aggregate)
__global__ void ema_chunk_sums(const float* __restrict__ x,
                               const float* __restrict__ al,
                               float* __restrict__ qf,
                               float* __restrict__ qb) {
    const int gid = blockIdx.x * blockDim.x + threadIdx.x;
    const int c = gid % Cc;
    const int j = (gid / Cc) % NCk;
    const int b = gid / (Cc * NCk);

    const float a = sigmoid_f(al[c]);
    const float w = 1.0f - a;

    const float* p = x + ((size_t)(b * Tt + j * Lc)) * Cc + c;
    float Qf = 0.0f, Qb = 0.0f, pw = a;
    #pragma unroll 8
    for (int i = 0; i < Lc; ++i) {
        const float xi = p[(size_t)i * Cc];
        Qf = w * Qf + a * xi;   // ends as sum a*w^(L-1-s) x[s]
        Qb += pw * xi;          // sum a*w^s x[s]
        pw *= w;
    }
    const int idx = (b * NCk + j) * Cc + c;
    qf[idx] = Qf;
    qb[idx] = Qb;
}

// ---------------- Kernel 2: sequential carry recombine (in place) -----------
// Forward carry into chunk j: cf[0]=x[b,0,c]; cf[j]=w^L*cf[j-1]+Qf[j-1].
// Backward carry into chunk j (from right): cb[NC-1]=x[b,T-1,c];
//   cb[j]=w^L*cb[j+1]+Qb[j+1].  Overwrites qf/qb with the carries.
__global__ void ema_carry_scan(const float* __restrict__ x,
                               const float* __restrict__ al,
                               float* __restrict__ qf,
                               float* __restrict__ qb) {
    const int gid = blockIdx.x * blockDim.x + threadIdx.x;  // b*Cc + c
    const int c = gid % Cc;
    const int b = gid / Cc;

    const float a = sigmoid_f(al[c]);
    const float w = 1.0f - a;
    float wL = w;
    #pragma unroll
    for (int k = 0; k < 6; ++k) wL *= wL;   // w^64 (Lc = 2^6)

    float carry = x[(size_t)(b * Tt) * Cc + c];             // x[b,0,c]
    for (int j = 0; j < NCk; ++j) {
        const int idx = (b * NCk + j) * Cc + c;
        const float t = qf[idx];
        qf[idx] = carry;
        carry = wL * carry + t;
    }
    carry = x[((size_t)(b * Tt) + Tt - 1) * Cc + c];        // x[b,T-1,c]
    for (int j = NCk - 1; j >= 0; --j) {
        const int idx = (b * NCk + j) * Cc + c;
        const float t = qb[idx];
        qb[idx] = carry;
        carry = wL * carry + t;
    }
}

// ---------------- Fallback (if workspace too small): row-wise 2-pass --------
__global__ void ema_rowwise(const float* __restrict__ x,
                            const float* __restrict__ al,
                            float* __restrict__ out) {
    const int gid = blockIdx.x * blockDim.x + threadIdx.x;  // b*Cc + c
    const int c = gid % Cc;
    const int b = gid / Cc;
    const float a = sigmoid_f(al[c]);
    const float w = 1.0f - a;

    const float* xr = x + (size_t)b * Tt * Cc + c;
    float* orow = out + (size_t)b * Tt * Cc + c;

    float g = xr[(size_t)(Tt - 1) * Cc];
    orow[(size_t)(Tt - 1) * Cc] = 0.5f * g;
    for (int t = Tt - 2; t >= 0; --t) {
        g = w * g + a * xr[(size_t)t * Cc];
        orow[(size_t)t * Cc] = 0.5f * g;
    }
    float f = xr[0];
    orow[0] += 0.5f * f;
    for (int t = 1; t < Tt; ++t) {
        f = w * f + a * xr[(size_t)t * Cc];
        orow[(size_t)t * Cc] += 0.5f * f;
    }
}

extern "C" void kernel_launch(void* const* d_in, const int* in_sizes, int n_in,
                              void* d_out, int out_size, void* d_ws, size_t ws_size,
                              hipStream_t stream) {
    (void)in_sizes; (void)n_in; (void)out_size;
    const float* x  = (const float*)d_in[0];
    const float* al = (const float*)d_in[1];
    float* out = (float*)d_out;

    const size_t nQ = (size_t)Bb * Cc * NCk;            // 524,288 floats
    const size_t need = 2 * nQ * sizeof(float);         // 4 MB

    if (d_ws && ws_size >= need) {
        float* qf = (float*)d_ws;
        float* qb = qf + nQ;
        ema_chunk_sums<<<(int)(nQ / 256), 256, 0, stream>>>(x, al, qf, qb);
        ema_carry_scan<<<(Bb * Cc) / 256, 256, 0, stream>>>(x, al, qf, qb);
        ema_fill<<<(int)(nQ / 128), 128, 0, stream>>>(x, al, qf, qb, out);
    } else {
        ema_rowwise<<<(Bb * Cc) / 256, 256, 0, stream>>>(x, al, out);
    }
}